// YOLOLoss_63453846831681
// MI455X (gfx1250) — compile-verified
//
#include <hip/hip_runtime.h>
#include <math.h>

// ---------------------------------------------------------------------------
// YOLO loss for MI455X (gfx1250, wave32).
// Memory-bound: only channels 0..4 are read for all 277k cells; the 80 class
// channels are read only at the <=320 obj-masked cells. Target table staged
// into LDS with CDNA5 async global->LDS copy; block reduction uses
// V_WMMA_F32_16X16X4_F32 (partials x ones == exact row sums).
// ---------------------------------------------------------------------------

typedef float v2f __attribute__((ext_vector_type(2)));
typedef float v8f __attribute__((ext_vector_type(8)));

#ifndef __has_builtin
#define __has_builtin(x) 0
#endif

typedef __attribute__((address_space(1))) int gas_int;   // global (AS1)
typedef __attribute__((address_space(3))) int las_int;   // LDS (AS3)

#define NB_B 16
#define NB_T 20
#define NB_A 3
#define NB_C 80
#define GWID 76
#define GHEI 76
#define PLANE (GWID * GHEI)            // 5776
#define CELLS_PER_B (NB_A * PLANE)     // 17328
#define PREP_STRIDE 12
#define TBL_FLOATS (NB_T * PREP_STRIDE)  // 240

__device__ __forceinline__ float sigmf(float x) { return 1.0f / (1.0f + expf(-x)); }
// jnp.clip(log(p), -100.0): lower clip only (handles -inf from log(0))
__device__ __forceinline__ float clog(float p) { return fmaxf(logf(p), -100.0f); }

__global__ void zero_kernel(float* out, int n) {
  int i = blockIdx.x * blockDim.x + threadIdx.x;
  if (i < n) out[i] = 0.0f;
}

// One thread per (b, t): precompute grid-space GT box, best anchor, cell, class.
__global__ void prep_targets_kernel(const float* __restrict__ txywh,
                                    const int* __restrict__ tcls,
                                    float* __restrict__ prep) {
  int i = blockIdx.x * blockDim.x + threadIdx.x;
  if (i >= NB_B * NB_T) return;
  float tx = txywh[i * 4 + 0];
  float ty = txywh[i * 4 + 1];
  float tw = txywh[i * 4 + 2];
  float th = txywh[i * 4 + 3];
  float gx = tx * (float)GWID;
  float gy = ty * (float)GHEI;
  float gw = tw * (float)GWID;
  float gh = th * (float)GHEI;
  int gi = (int)floorf(gx);
  int gj = (int)floorf(gy);
  // ANCHORS / IMG(608)
  const float SAW[9] = {12.f/608.f, 19.f/608.f, 40.f/608.f, 36.f/608.f, 76.f/608.f,
                        72.f/608.f, 142.f/608.f, 192.f/608.f, 459.f/608.f};
  const float SAH[9] = {16.f/608.f, 36.f/608.f, 28.f/608.f, 75.f/608.f, 55.f/608.f,
                        146.f/608.f, 110.f/608.f, 243.f/608.f, 401.f/608.f};
  float best = -1.0f;
  int bi = 0;
  for (int k = 0; k < 9; ++k) {
    float inter = fminf(tw, SAW[k]) * fminf(th, SAH[k]);
    float iou = inter / (tw * th + SAW[k] * SAH[k] - inter);
    if (iou > best) { best = iou; bi = k; }   // first max wins (argmax semantics)
  }
  int local = bi - 6;                         // head_id=2 (W=76) -> anc0 = 6
  int aidx = (local >= 0 && local < NB_A) ? local : NB_A;  // NB_A == dropped
  float* o = prep + (size_t)i * PREP_STRIDE;
  o[0] = gx;  o[1] = gy;  o[2] = gw;  o[3] = gh;
  o[4] = (float)aidx; o[5] = (float)gi; o[6] = (float)gj;
  o[7] = (float)tcls[i]; o[8] = tw * th;
  o[9] = 0.0f; o[10] = 0.0f; o[11] = 0.0f;
}

__global__ void __launch_bounds__(256)
yolo_loss_kernel(const float* __restrict__ head,
                 const float* __restrict__ prep,
                 float* __restrict__ out) {
  __shared__ float tdata[TBL_FLOATS];
  __shared__ float wsum[8];
  const int tid = threadIdx.x;
  const int b = blockIdx.y;

  // --- Stage this batch's 240-float target table into LDS (async path) ---
#if __has_builtin(__builtin_amdgcn_global_load_async_to_lds_b32)
  if (tid < TBL_FLOATS) {
    gas_int* g = (gas_int*)(prep + (size_t)b * TBL_FLOATS + tid);
    las_int* l = (las_int*)&tdata[tid];
    __builtin_amdgcn_global_load_async_to_lds_b32(g, l, 0, 0);
  }
#if __has_builtin(__builtin_amdgcn_s_wait_asynccnt)
  __builtin_amdgcn_s_wait_asynccnt(0);
#else
  asm volatile("s_wait_asynccnt 0" ::: "memory");
#endif
#else
  if (tid < TBL_FLOATS) tdata[tid] = prep[(size_t)b * TBL_FLOATS + tid];
#endif
  __syncthreads();

  const int idx = blockIdx.x * blockDim.x + tid;
  float acc = 0.0f;
  if (idx < CELLS_PER_B) {
    const int a = idx / PLANE;
    const int rem = idx - a * PLANE;
    const int hh = rem / GWID;
    const int ww = rem - hh * GWID;
    // inference[b,a,h,w,c] = head[b, a*85 + c, h, w]
    const float* base = head + ((size_t)b * (NB_A * (5 + NB_C)) + (size_t)a * (5 + NB_C)) * PLANE + rem;
    const float in0 = base[0];
    const float in1 = base[PLANE];
    const float in2 = base[2 * PLANE];
    const float in3 = base[3 * PLANE];
    const float in4 = base[4 * PLANE];
    const float AW[3] = {142.f/8.f, 192.f/8.f, 459.f/8.f};   // head 2 anchors / grid
    const float AH[3] = {110.f/8.f, 243.f/8.f, 401.f/8.f};
    const float xx = 1.1f * sigmf(in0) - 0.05f;              // EGS = 0.1
    const float yy = 1.1f * sigmf(in1) - 0.05f;
    const float pw = expf(in2) * AW[a];
    const float ph = expf(in3) * AH[a];
    const float px = xx + (float)ww;
    const float py = yy + (float)hh;
    const float ap = (pw - px) * (ph - py);  // replicate reference's corner-convention area

    float maxiou = -INFINITY;
    int tsel = -1;
    for (int t = 0; t < NB_T; ++t) {
      const float* td = &tdata[t * PREP_STRIDE];
      const float g0 = td[0], g1 = td[1], g2 = td[2], g3 = td[3];
      const float lt0 = fmaxf(g0, px), lt1 = fmaxf(g1, py);
      const float rb0 = fminf(g2, pw), rb1 = fminf(g3, ph);
      const float wi0 = fmaxf(rb0 - lt0, 0.0f), wi1 = fmaxf(rb1 - lt1, 0.0f);
      const float inter = wi0 * wi1;
      const float ag = (g2 - g0) * (g3 - g1);
      const float iou = inter / (ag + ap - inter);
      maxiou = fmaxf(maxiou, iou);
      if ((int)td[4] == a && (int)td[5] == ww && (int)td[6] == hh) tsel = t;  // last set wins
    }
    const bool m = (tsel >= 0);
    const float sc = sigmf(in4);
    if (m) {
      // conf BCE with target 1 (noobj contribution is zero at obj cells)
      acc += -clog(sc);
      const float* td = &tdata[tsel * PREP_STRIDE];
      const float gx = td[0], gy = td[1], gw = td[2], gh = td[3];
      // CIoU (center/size convention)
      const float b1nx = px - 0.5f * pw, b1mx = px + 0.5f * pw;
      const float b1ny = py - 0.5f * ph, b1my = py + 0.5f * ph;
      const float b2nx = gx - 0.5f * gw, b2mx = gx + 0.5f * gw;
      const float b2ny = gy - 0.5f * gh, b2my = gy + 0.5f * gh;
      const float iwx = fmaxf(fminf(b1mx, b2mx) - fmaxf(b1nx, b2nx), 0.0f);
      const float iwy = fmaxf(fminf(b1my, b2my) - fmaxf(b1ny, b2ny), 0.0f);
      const float ia = iwx * iwy;
      const float a1 = pw * ph, a2 = gw * gh;
      const float iou = ia / (a1 + a2 - ia);
      const float dx = px - gx, dy = py - gy;
      const float csq = dx * dx + dy * dy;
      const float ewx = fmaxf(b1mx, b2mx) - fminf(b1nx, b2nx);
      const float ewy = fmaxf(b1my, b2my) - fminf(b1ny, b2ny);
      const float dsq = ewx * ewx + ewy * ewy;
      const float diou = iou - csq / dsq;
      const float dv = atanf(pw / ph) - atanf(gw / gh);
      const float v = 0.40528473456935109f * dv * dv;  // 4/pi^2
      const float ciou = diou - v / (1.0f - iou + v) * v;
      acc += (1.0f - ciou) * (2.0f - td[8]);
      // class BCE (only at obj cells: <=320 total, so the 80 extra channel
      // reads stay off the critical bandwidth path)
      const int cls = (int)td[7];
      for (int c = 0; c < NB_C; ++c) {
        const float p = sigmf(base[(size_t)(5 + c) * PLANE]);
        acc += (c == cls) ? -clog(p) : -clog(1.0f - p);
      }
    } else {
      const float noobj = (maxiou > 0.5f) ? 0.0f : 1.0f;
      acc += noobj * (-clog(1.0f - sc));
    }
  }

  // --- Wave reduction via V_WMMA_F32_16X16X4_F32: A = partials (16x4 with
  // zero K=1/K=3 columns), B = ones(4x16) -> D[m][n] = p_m + p_{m+16}.
  // Lane n (0..15) holds D[0..7][n]; lane n+16 holds D[8..15][n]. EXEC is
  // all-ones here: no thread returned early.
  float rsum;
#if __has_builtin(__builtin_amdgcn_wmma_f32_16x16x4_f32)
  v2f Av; Av.x = acc;  Av.y = 0.0f;
  v2f Bv; Bv.x = 1.0f; Bv.y = 1.0f;
  v8f Cz = {0.f, 0.f, 0.f, 0.f, 0.f, 0.f, 0.f, 0.f};
  v8f D = __builtin_amdgcn_wmma_f32_16x16x4_f32(false, Av, false, Bv,
                                                (short)0, Cz, false, false);
  float r = D[0] + D[1] + D[2] + D[3] + D[4] + D[5] + D[6] + D[7];
  rsum = __shfl(r, 0, 32) + __shfl(r, 16, 32);
#else
  rsum = acc;
  for (int off = 16; off >= 1; off >>= 1) rsum += __shfl_xor(rsum, off, 32);
#endif
  const int lane = tid & 31;
  const int wid = tid >> 5;
  if (lane == 0) wsum[wid] = rsum;
  __syncthreads();
  if (tid == 0) {
    float bs = wsum[0] + wsum[1] + wsum[2] + wsum[3] +
               wsum[4] + wsum[5] + wsum[6] + wsum[7];
    atomicAdd(out, bs);
  }
}

extern "C" void kernel_launch(void* const* d_in, const int* in_sizes, int n_in,
                              void* d_out, int out_size, void* d_ws, size_t ws_size,
                              hipStream_t stream) {
  (void)in_sizes; (void)n_in; (void)ws_size;
  const float* head  = (const float*)d_in[0];
  const float* txywh = (const float*)d_in[1];
  const int*   tcls  = (const int*)d_in[2];
  float* out  = (float*)d_out;
  float* prep = (float*)d_ws;  // 16*20*12 floats = 15360 B

  zero_kernel<<<1, 64, 0, stream>>>(out, out_size);
  prep_targets_kernel<<<1, NB_B * NB_T, 0, stream>>>(txywh, tcls, prep);
  dim3 grid((CELLS_PER_B + 255) / 256, NB_B, 1);
  yolo_loss_kernel<<<grid, 256, 0, stream>>>(head, prep, out);
}